// EnhancedGradientConsistencyLoss_72541997629798
// MI455X (gfx1250) — compile-verified
//
#include <hip/hip_runtime.h>
#include <math.h>

typedef float v2f __attribute__((ext_vector_type(2)));
typedef float v8f __attribute__((ext_vector_type(8)));

#define Bn 8
#define Cn 3
#define Hn 512
#define Wn 512
#define NTOT (Bn*Cn*Hn*Wn)            // 6291456
#define NPLANES (Bn*Cn)               // 24
#define NTILES (NPLANES*32*32)        // 24576 (16x16 tiles)
#define WAVES_PER_BLOCK 8
#define NBLOCKS (NTILES/WAVES_PER_BLOCK)  // 3072

// Per-wave LDS layout (floats)
#define MOFF 0            // 24x24 mask tile, stride 25 -> 600
#define YOFF 600          // 16x24 vert-filtered, stride 25 -> 400
#define OOFF 1000         // 18x18 output tile, stride 19 -> 342
#define TOFF 1342         // 18x18 target tile, stride 19 -> 342
#define WSTRIDE 1696

// Gaussian taps: sigma=1, radius=4, normalized (compile-time only; all uses
// are either constant-folded literals or staged into the LDS band table).
__device__ constexpr float K9c[9] = {
    1.3383062e-04f, 4.4318617e-03f, 5.3991124e-02f, 2.4197145e-01f,
    3.9894349e-01f,
    2.4197145e-01f, 5.3991124e-02f, 4.4318617e-03f, 1.3383062e-04f
};

__global__ __launch_bounds__(256)
void egc_main(const float* __restrict__ outp, const float* __restrict__ tgtp,
              const int* __restrict__ maskp, float* __restrict__ partial)
{
    __shared__ float lds[WAVES_PER_BLOCK * WSTRIDE];
    __shared__ float blkred[WAVES_PER_BLOCK][4];
    __shared__ float k9tab[48];   // zero-padded band: k9tab[16+d]=K9c[d], d in [0,8]

    const int tid  = threadIdx.x;
    const int wave = tid >> 5;
    const int lane = tid & 31;

    // Build the band table from literals (cndmask chain; no memory-indexed
    // constant access -> no divergent global loads near the WMMAs).
    if (tid < 48) {
        int d = tid - 16;
        float v = 0.0f;
        #pragma unroll
        for (int t = 0; t < 9; ++t) v = (d == t) ? K9c[t] : v;
        k9tab[tid] = v;
    }

    float* sm  = &lds[wave * WSTRIDE + MOFF];
    float* syv = &lds[wave * WSTRIDE + YOFF];
    float* so  = &lds[wave * WSTRIDE + OOFF];
    float* st  = &lds[wave * WSTRIDE + TOFF];

    const int tile  = blockIdx.x * WAVES_PER_BLOCK + wave;
    const int plane = tile >> 10;          // / (32*32)
    const int trow  = (tile & 1023) >> 5;
    const int tcol  = tile & 31;
    const int r0 = trow << 4;
    const int c0 = tcol << 4;
    const size_t pbase = (size_t)plane * (size_t)(Hn * Wn);

    // ---- Stage mask tile 24x24 with symmetric reflection (jnp 'symmetric')
    for (int e = lane; e < 576; e += 32) {
        int rr = e / 24, cc = e - rr * 24;
        int gr = r0 - 4 + rr; gr = gr < 0 ? -1 - gr : (gr >= Hn ? 2 * Hn - 1 - gr : gr);
        int gc = c0 - 4 + cc; gc = gc < 0 ? -1 - gc : (gc >= Wn ? 2 * Wn - 1 - gc : gc);
        sm[rr * 25 + cc] = (float)maskp[pbase + (size_t)gr * Wn + gc];
    }
    // ---- Stage output/target 18x18 tiles with zero padding (Sobel halo)
    for (int e = lane; e < 324; e += 32) {
        int rr = e / 18, cc = e - rr * 18;
        int gr = r0 - 1 + rr, gc = c0 - 1 + cc;
        bool in = (gr >= 0) & (gr < Hn) & (gc >= 0) & (gc < Wn);
        size_t gi = pbase + (size_t)gr * Wn + gc;
        so[rr * 19 + cc] = in ? outp[gi] : 0.0f;
        st[rr * 19 + cc] = in ? tgtp[gi] : 0.0f;
    }
    __syncthreads();

    // ---- Vertical Gaussian: 24x24 -> 16x24 (VALU; taps fold to literals)
    for (int e = lane; e < 384; e += 32) {
        int rr = e / 24, cc = e - rr * 24;
        float s = 0.0f;
        #pragma unroll
        for (int t = 0; t < 9; ++t) s = fmaf(K9c[t], sm[(rr + t) * 25 + cc], s);
        syv[rr * 25 + cc] = s;
    }
    __syncthreads();

    // ---- Horizontal Gaussian as banded matmul via V_WMMA_F32_16X16X4_F32
    // Y(16x16) = Xv(16x24) * W(24x16), W[i][j] = K9c[i-j] for 0<=i-j<=8.
    // f32 A(16x4): lanes 0-15 hold K={0,1}, lanes 16-31 hold K={2,3}; row M = lane&15.
    // f32 B(4x16): lanes 0-15 hold rows {0,1}, lanes 16-31 rows {2,3}; col N = lane&15.
    const int n    = lane & 15;
    const int half = lane >> 4;
    const float* bt = &k9tab[16 - n];   // bt[i] = W[i][n], zero-padded band
    v8f acc = {0.f, 0.f, 0.f, 0.f, 0.f, 0.f, 0.f, 0.f};
    #pragma unroll
    for (int ch = 0; ch < 6; ++ch) {
        const int kb = ch * 4 + half * 2;   // this lane's two K indices: kb, kb+1
        v2f a, b;
        a.x = syv[n * 25 + kb];
        a.y = syv[n * 25 + kb + 1];
        b.x = bt[kb];                       // unconditional ds_load (band is padded)
        b.y = bt[kb + 1];
        acc = __builtin_amdgcn_wmma_f32_16x16x4_f32(
                  false, a, false, b, (short)0, acc, false, false);
    }

    // D layout: VGPR j, lanes 0-15 -> row j, col=lane; lanes 16-31 -> row j+8, col=lane-16.
    // => this lane owns pixels (R = half*8 + j, col = n), j = 0..7.
    float s_mag = 0.0f, s_dir = 0.0f, s_w = 0.0f;
    const int Rb = half * 8;
    float o00,o01,o02,o10,o11,o12,o20,o21,o22;
    float t00,t01,t02,t10,t11,t12,t20,t21,t22;
    {
        const float* p = so + Rb * 19 + n;   // rows Rb..Rb+2 (array), cols n..n+2
        o00=p[0];  o01=p[1];  o02=p[2];
        o10=p[19]; o11=p[20]; o12=p[21];
        o20=p[38]; o21=p[39]; o22=p[40];
        const float* q = st + Rb * 19 + n;
        t00=q[0];  t01=q[1];  t02=q[2];
        t10=q[19]; t11=q[20]; t12=q[21];
        t20=q[38]; t21=q[39]; t22=q[40];
    }
    #pragma unroll
    for (int j = 0; j < 8; ++j) {
        float smv = fminf(fmaxf(acc[j], 0.0f), 1.0f);
        float w   = fminf(fmaxf(1.0f - fabsf(smv - 0.5f) * 2.0f, 0.0f), 1.0f);

        float gx = (o02 + 2.0f*o12 + o22) - (o00 + 2.0f*o10 + o20);
        float gy = (o20 + 2.0f*o21 + o22) - (o00 + 2.0f*o01 + o02);
        float mo = sqrtf(gx*gx + gy*gy + 1e-8f);
        float po = atan2f(gy, gx);

        float hx = (t02 + 2.0f*t12 + t22) - (t00 + 2.0f*t10 + t20);
        float hy = (t20 + 2.0f*t21 + t22) - (t00 + 2.0f*t01 + t02);
        float mt = sqrtf(hx*hx + hy*hy + 1e-8f);
        float pt = atan2f(hy, hx);

        s_mag += fabsf(mo * w - mt * w);
        float dd = fabsf(po - pt);
        dd = fminf(dd, 6.28318530717958647692f - dd);
        s_dir += dd * w;
        s_w   += w;

        if (j < 7) {  // roll 3x3 window down one row
            o00=o10; o01=o11; o02=o12;  o10=o20; o11=o21; o12=o22;
            t00=t10; t01=t11; t02=t12;  t10=t20; t11=t21; t12=t22;
            const float* p = so + (Rb + j + 3) * 19 + n;
            o20=p[0]; o21=p[1]; o22=p[2];
            const float* q = st + (Rb + j + 3) * 19 + n;
            t20=q[0]; t21=q[1]; t22=q[2];
        }
    }

    // ---- Deterministic reduction: wave32 shfl tree -> per-block partial triple
    #pragma unroll
    for (int off = 16; off > 0; off >>= 1) {
        s_mag += __shfl_xor(s_mag, off, 32);
        s_dir += __shfl_xor(s_dir, off, 32);
        s_w   += __shfl_xor(s_w,   off, 32);
    }
    if (lane == 0) {
        blkred[wave][0] = s_mag; blkred[wave][1] = s_dir; blkred[wave][2] = s_w;
    }
    __syncthreads();
    if (tid == 0) {
        float a = 0.f, b = 0.f, c = 0.f;
        #pragma unroll
        for (int wv = 0; wv < WAVES_PER_BLOCK; ++wv) {
            a += blkred[wv][0]; b += blkred[wv][1]; c += blkred[wv][2];
        }
        partial[blockIdx.x * 3 + 0] = a;
        partial[blockIdx.x * 3 + 1] = b;
        partial[blockIdx.x * 3 + 2] = c;
    }
}

__global__ __launch_bounds__(256)
void egc_final(const float* __restrict__ partial, float* __restrict__ outv)
{
    __shared__ float r0[256], r1[256], r2[256];
    const int t = threadIdx.x;
    float a = 0.f, b = 0.f, c = 0.f;
    for (int i = t; i < NBLOCKS; i += 256) {     // fixed order -> deterministic
        a += partial[i * 3 + 0];
        b += partial[i * 3 + 1];
        c += partial[i * 3 + 2];
    }
    r0[t] = a; r1[t] = b; r2[t] = c;
    __syncthreads();
    for (int s = 128; s > 0; s >>= 1) {
        if (t < s) { r0[t] += r0[t + s]; r1[t] += r1[t + s]; r2[t] += r2[t + s]; }
        __syncthreads();
    }
    if (t == 0) {
        float Smag = r0[0], Sdir = r1[0], Sw = r2[0];
        const float N = (float)NTOT;
        float mag_loss = Smag / N;
        if (Sw > 0.0f) mag_loss = mag_loss / (Sw / N + 1e-8f);
        float dir_loss = (Sw > 0.0f) ? Sdir / (Sw + 1e-8f) : Sdir;
        outv[0] = mag_loss + dir_loss;
    }
}

extern "C" void kernel_launch(void* const* d_in, const int* in_sizes, int n_in,
                              void* d_out, int out_size, void* d_ws, size_t ws_size,
                              hipStream_t stream)
{
    const float* outp = (const float*)d_in[0];
    const float* tgtp = (const float*)d_in[1];
    const int*   mskp = (const int*)d_in[2];
    float* partial = (float*)d_ws;   // NBLOCKS*3 floats, fully overwritten every call

    egc_main<<<NBLOCKS, 256, 0, stream>>>(outp, tgtp, mskp, partial);
    egc_final<<<1, 256, 0, stream>>>(partial, (float*)d_out);
}